// MaterialModelFinder_4913442586755
// MI455X (gfx1250) — compile-verified
//
#include <hip/hip_runtime.h>
#include <stdint.h>

typedef float v4f __attribute__((ext_vector_type(4)));
typedef int   v4i __attribute__((ext_vector_type(4)));

// AS-qualified pointee types for the async-to-LDS builtin
typedef __attribute__((address_space(1))) v4i g_v4i;   // global
typedef __attribute__((address_space(3))) v4i l_v4i;   // LDS

#define N_GAUSS 8
#define PAIRS_PER_BLOCK 256   // 256 threads, one t-pair each -> 512 t values / block
#define LDS_BYTES (PAIRS_PER_BLOCK * 2 * 4)  // 2048 B of disp row

// state[e,g,t,c] = (p+p^2) * disp[e,t] * (g+1), broadcast over c=0..5.
// Store-bandwidth bound: 1.61 GB out vs 32 MB in. Strategy:
//   - async-DMA the 2 KB disp slice into LDS (gfx1250 global_load_async_to_lds_b128)
//   - each thread emits 3x b128 non-temporal stores per gauss point
__global__ void __launch_bounds__(256)
state_hist_kernel(const float* __restrict__ disp,
                  const float* __restrict__ param,
                  float* __restrict__ out,
                  int n_time) {
    __shared__ float ldsDisp[PAIRS_PER_BLOCK * 2];

    const int tid   = threadIdx.x;        // 0..255
    const int chunk = blockIdx.x;         // which 512-t slice
    const int e     = blockIdx.y;         // element

    // ---- Stage 2 KB of disp[e, chunk*512 : chunk*512+512] into LDS ----
    const char* gsrc = (const char*)(disp + (size_t)e * n_time + (size_t)chunk * (PAIRS_PER_BLOCK * 2));

#if defined(__has_builtin) && __has_builtin(__builtin_amdgcn_global_load_async_to_lds_b128)
    if (tid < LDS_BYTES / 16) {
        // integer round-trip to materialize explicit-AS pointers
        // (flat LDS addr low 32 bits == LDS offset per ISA 10.2)
        uintptr_t gaddr = (uintptr_t)(gsrc + tid * 16);
        uint32_t  loff  = (uint32_t)(uintptr_t)((char*)ldsDisp + tid * 16);
        __builtin_amdgcn_global_load_async_to_lds_b128(
            (g_v4i*)gaddr,
            (l_v4i*)(uintptr_t)loff,
            0, 0);
    }
  #if __has_builtin(__builtin_amdgcn_s_wait_asynccnt)
    __builtin_amdgcn_s_wait_asynccnt(0);
  #else
    asm volatile("s_wait_asynccnt 0" ::: "memory");
  #endif
#else
    // Fallback: cooperative vector loads into LDS
    if (tid < LDS_BYTES / 16) {
        v4f tmp = *(const v4f*)(gsrc + tid * 16);
        *(v4f*)((char*)ldsDisp + tid * 16) = tmp;
    }
#endif
    __syncthreads();

    // ---- Per-thread values ----
    const float p    = param[0];
    const float coef = p + p * p;

    const float d0 = ldsDisp[2 * tid + 0];
    const float d1 = ldsDisp[2 * tid + 1];

    const int    t0     = 2 * (chunk * PAIRS_PER_BLOCK + tid);     // first t of this pair
    const size_t rowLen = (size_t)n_time * 6;                      // floats per (e,g) row
    float* base = out + (size_t)e * N_GAUSS * rowLen + (size_t)t0 * 6;  // 48B-aligned

    #pragma unroll
    for (int g = 0; g < N_GAUSS; ++g) {
        const float s  = coef * (float)(g + 1);
        const float v0 = s * d0;
        const float v1 = s * d1;
        v4f a = {v0, v0, v0, v0};
        v4f b = {v0, v0, v1, v1};
        v4f c = {v1, v1, v1, v1};
        v4f* p4 = (v4f*)(base + (size_t)g * rowLen);
        __builtin_nontemporal_store(a, p4 + 0);
        __builtin_nontemporal_store(b, p4 + 1);
        __builtin_nontemporal_store(c, p4 + 2);
    }
}

extern "C" void kernel_launch(void* const* d_in, const int* in_sizes, int n_in,
                              void* d_out, int out_size, void* d_ws, size_t ws_size,
                              hipStream_t stream) {
    const float* disp  = (const float*)d_in[0];   // (n_elem, n_time) f32
    // d_in[1] = time_hist: unused by the reference math
    const float* param = (const float*)d_in[2];   // (1,) f32
    float* out = (float*)d_out;                   // (n_elem, 8, n_time, 6) f32

    const int n_time = in_sizes[1];               // 4096
    const int n_elem = in_sizes[0] / n_time;      // 2048

    const int tvals_per_block = PAIRS_PER_BLOCK * 2;   // 512
    dim3 grid(n_time / tvals_per_block, n_elem);       // (8, 2048)
    dim3 block(PAIRS_PER_BLOCK);                       // 256 = 8 wave32

    state_hist_kernel<<<grid, block, 0, stream>>>(disp, param, out, n_time);
}